// EquivariantProductBasisWithOneBodySelfMagmomBlock_42125039239739
// MI455X (gfx1250) — compile-verified
//
#include <hip/hip_runtime.h>

#define BLOCK 128

typedef __attribute__((ext_vector_type(2))) float v2f;
typedef __attribute__((ext_vector_type(8))) float v8f;

#define V8F_ZERO {0.f, 0.f, 0.f, 0.f, 0.f, 0.f, 0.f, 0.f}

__device__ __forceinline__ v8f wmma4(v8f acc, v2f a, v2f b) {
  // D = A(16x4,f32) * B(4x16,f32) + C(16x16,f32)
  return __builtin_amdgcn_wmma_f32_16x16x4_f32(false, a, false, b, (short)0, acc,
                                               false, false);
}

__device__ __forceinline__ float fast_silu(float x) {
  // x * sigmoid(x), with v_rcp_f32 instead of precise division
  return x * __builtin_amdgcn_rcpf(1.f + __expf(-x));
}

// Preload A (16 x K, LDS) into registers: K/4 v2f per lane.
template <int K>
__device__ __forceinline__ void load_areg(const float* A, int lda, int lane,
                                          v2f* ar) {
  const int n16 = lane & 15, kh = (lane >> 4) << 1;
#pragma unroll
  for (int k0 = 0; k0 < K; k0 += 4) {
    ar[k0 >> 2][0] = A[n16 * lda + k0 + kh];
    ar[k0 >> 2][1] = A[n16 * lda + k0 + kh + 1];
  }
}

// 16x16 tile accumulate with register-resident A, global B.
template <int K>
__device__ __forceinline__ v8f gemm_regA(v8f acc, const v2f* ar,
                                         const float* __restrict__ B, int ldb,
                                         int col0, int lane) {
  const int n16 = lane & 15, kh = (lane >> 4) << 1;
#pragma unroll
  for (int k0 = 0; k0 < K; k0 += 4) {
    const int kk = k0 + kh;
    v2f b;
    b[0] = B[kk * ldb + col0 + n16];
    b[1] = B[(kk + 1) * ldb + col0 + n16];
    acc = wmma4(acc, ar[k0 >> 2], b);
  }
  return acc;
}

// Hidden MLP layer: Hout = silu((Hin @ W) * scale); A hoisted to registers.
template <int K>
__device__ __forceinline__ void mlp_hidden(const float* Hin, int ldin,
                                           const float* __restrict__ W, int Cout,
                                           float scale, float* Hout, int wave,
                                           int lane) {
  v2f ar[K / 4];
  load_areg<K>(Hin, ldin, lane, ar);
  const int n16 = lane & 15, rh = lane >> 4;
  for (int j = wave; j < (Cout >> 4); j += 4) {
    v8f acc = V8F_ZERO;
    acc = gemm_regA<K>(acc, ar, W, Cout, j * 16, lane);
#pragma unroll
    for (int t = 0; t < 8; ++t) {
      float x = fast_silu(acc[t] * scale);
      Hout[(t + 8 * rh) * Cout + j * 16 + n16] = x;
    }
  }
}

__global__ __launch_bounds__(BLOCK) void magmom_fused(
    const float* __restrict__ s_in, const float* __restrict__ v_in,
    const float* __restrict__ sc, const float* __restrict__ attrs,
    const float* __restrict__ invf, const float* __restrict__ mattr,
    const float* __restrict__ mlen,
    const float* __restrict__ Ws, const float* __restrict__ Wv,
    const float* __restrict__ Wc1, const float* __restrict__ Wc2,
    const float* __restrict__ Wc3, const float* __restrict__ Wc4,
    const float* __restrict__ Wb1, const float* __restrict__ Wb2,
    const float* __restrict__ Wb3, const float* __restrict__ Wb4,
    const float* __restrict__ expsc,
    const float* __restrict__ Wl0, const float* __restrict__ Wl1,
    const float* __restrict__ Wo0, const float* __restrict__ Wo1,
    float* __restrict__ out) {
  extern __shared__ float smem[];
  float* sS    = smem;          // 16 x 128  : out_s
  float* sV    = sS + 2048;     // 3 x 16 x 128 : out_v planes
  float* sScal = sV + 6144;     // 16 x 256  : [m0 | m1]
  float* sW23  = sScal + 4096;  // 16 x 256  : [w2 | w3]
  float* sHA   = sW23 + 4096;   // 16 x 64 MLP ping
  float* sHB   = sHA + 1024;    // 16 x 64 MLP pong
  float* sInv  = sHB + 1024;    // 16 x 16 magmom inv feats
  float* sY    = sInv + 256;    // 16 x 4 magmom attrs
  float* sC0   = sY + 64;       // 16 : (1/sqrt2) * y0
  float* sF    = sC0 + 16;      // 16 : 1 - exp(-es*len)

  const int tid  = threadIdx.x;
  const int lane = tid & 31;
  const int wave = tid >> 5;
  const int base = blockIdx.x * 16;

  // ---- stage small per-node data ----
  for (int i = tid; i < 256; i += BLOCK) sInv[i] = invf[base * 16 + i];
  if (tid < 64) sY[tid] = mattr[base * 4 + tid];
  if (tid < 16) {
    sF[tid]  = 1.f - __expf(-expsc[0] * mlen[base + tid]);
    sC0[tid] = 0.70710678118654752f * mattr[(base + tid) * 4];
  }

  // ---- elementwise: out_s / out_v (one node per iteration, c = tid) ----
  for (int n = 0; n < 16; ++n) {
    const int c  = tid;  // BLOCK == C == 128
    const int gn = base + n;
    const float s = s_in[(size_t)gn * 128 + c];
    const float* vp = v_in + (size_t)(gn * 128 + c) * 3;
    const float v0 = vp[0], v1 = vp[1], v2 = vp[2];
    const float vv = v0 * v0 + v1 * v1 + v2 * v2;
    float ws0 = 0, ws1 = 0, ws2 = 0, ws3 = 0, ws4 = 0;
    float wv0 = 0, wv1 = 0, wv2 = 0, wv3 = 0;
    for (int e = 0; e < 10; ++e) {
      const float a = attrs[gn * 10 + e];  // one-hot: block-uniform branch
      if (a != 0.f) {
        const float* p = Ws + (e * 5) * 128 + c;
        ws0 += a * p[0];   ws1 += a * p[128]; ws2 += a * p[256];
        ws3 += a * p[384]; ws4 += a * p[512];
        const float* q = Wv + (e * 4) * 128 + c;
        wv0 += a * q[0];   wv1 += a * q[128];
        wv2 += a * q[256]; wv3 += a * q[384];
      }
    }
    const float s2 = s * s;
    const float os = ws0 * s + ws1 * s2 + ws2 * vv + ws3 * s2 * s + ws4 * s * vv;
    const float g  = wv0 + wv1 * s + wv2 * s2 + wv3 * vv;
    sS[n * 128 + c] = os;
    sV[(0 * 16 + n) * 128 + c] = g * v0;
    sV[(1 * 16 + n) * 128 + c] = g * v1;
    sV[(2 * 16 + n) * 128 + c] = g * v2;
  }
  __syncthreads();

  const int n16 = lane & 15, rh = lane >> 4, kh = rh << 1;

  // ---- ob MLP: 16->64->64->64->128; fuse f*ob into sScal[:,0:128] ----
  mlp_hidden<16>(sInv, 16, Wb1, 64, 0.25f, sHA, wave, lane);
  __syncthreads();
  mlp_hidden<64>(sHA, 64, Wb2, 64, 0.125f, sHB, wave, lane);
  __syncthreads();
  mlp_hidden<64>(sHB, 64, Wb3, 64, 0.125f, sHA, wave, lane);
  __syncthreads();
  {
    v2f ar[16];
    load_areg<64>(sHA, 64, lane, ar);
    for (int j = wave; j < 8; j += 4) {
      v8f acc = V8F_ZERO;
      acc = gemm_regA<64>(acc, ar, Wb4, 128, j * 16, lane);
#pragma unroll
      for (int t = 0; t < 8; ++t) {
        const int m = t + 8 * rh;
        sScal[m * 256 + j * 16 + n16] = sF[m] * (acc[t] * 0.125f);
      }
    }
  }
  __syncthreads();

  // ---- tp_w MLP: 16->64->64->64->512; fuse m0/m1 build + stash w2,w3 ----
  mlp_hidden<16>(sInv, 16, Wc1, 64, 0.25f, sHA, wave, lane);
  __syncthreads();
  mlp_hidden<64>(sHA, 64, Wc2, 64, 0.125f, sHB, wave, lane);
  __syncthreads();
  mlp_hidden<64>(sHB, 64, Wc3, 64, 0.125f, sHA, wave, lane);
  __syncthreads();
  {
    v2f ar[16];
    load_areg<64>(sHA, 64, lane, ar);
    for (int j = wave; j < 32; j += 4) {
      v8f acc = V8F_ZERO;
      acc = gemm_regA<64>(acc, ar, Wc4, 512, j * 16, lane);
      const int cg = j * 16 + n16;  // column in [0,512); region uniform per j
#pragma unroll
      for (int t = 0; t < 8; ++t) {
        const int m = t + 8 * rh;
        const float x = acc[t] * 0.125f;
        if (cg < 128) {  // m0 += INV_SQ2 * w0 * out_s * y0
          sScal[m * 256 + cg] +=
              0.70710678118654752f * x * sS[m * 128 + cg] * sY[m * 4];
        } else if (cg < 256) {  // m1 = INV_SQ6 * w1 * (out_v . y1)
          const int c = cg - 128;
          const float dot = sV[(0 * 16 + m) * 128 + c] * sY[m * 4 + 1] +
                            sV[(1 * 16 + m) * 128 + c] * sY[m * 4 + 2] +
                            sV[(2 * 16 + m) * 128 + c] * sY[m * 4 + 3];
          sScal[m * 256 + cg] = 0.40824829046386302f * x * dot;
        } else if (cg < 384) {  // raw w2
          sW23[m * 256 + (cg - 256)] = x;
        } else {  // raw w3
          sW23[m * 256 + 128 + (cg - 384)] = x;
        }
      }
    }
  }
  __syncthreads();

  // ---- output GEMMs: 8 msg0 tiles + 24 msg1 tiles across 4 waves ----
  const float inv2c = 0.0625f;               // 1/sqrt(256)
  const float invc  = 0.08838834764831845f;  // 1/sqrt(128)
  for (int T = wave; T < 32; T += 4) {
    if (T < 8) {
      const int j = T, cb = j * 16 + n16;
      v8f a1 = V8F_ZERO, a2 = V8F_ZERO;
      // Two independent WMMA chains interleaved: scal@Wl0 and out_s@Wo0.
#pragma unroll 4
      for (int k0 = 0; k0 < 128; k0 += 4) {
        const int kk = k0 + kh;
        v2f a, b;
        a[0] = sScal[n16 * 256 + kk];
        a[1] = sScal[n16 * 256 + kk + 1];
        b[0] = Wl0[kk * 128 + cb];
        b[1] = Wl0[(kk + 1) * 128 + cb];
        a1 = wmma4(a1, a, b);
        a[0] = sS[n16 * 128 + kk];
        a[1] = sS[n16 * 128 + kk + 1];
        b[0] = Wo0[kk * 128 + cb];
        b[1] = Wo0[(kk + 1) * 128 + cb];
        a2 = wmma4(a2, a, b);
      }
#pragma unroll 8
      for (int k0 = 128; k0 < 256; k0 += 4) {
        const int kk = k0 + kh;
        v2f a, b;
        a[0] = sScal[n16 * 256 + kk];
        a[1] = sScal[n16 * 256 + kk + 1];
        b[0] = Wl0[kk * 128 + cb];
        b[1] = Wl0[(kk + 1) * 128 + cb];
        a1 = wmma4(a1, a, b);
      }
#pragma unroll
      for (int t = 0; t < 8; ++t) {
        const int m = t + 8 * rh, gn = base + m, col = j * 16 + n16;
        out[(size_t)gn * 512 + col] =
            inv2c * a1[t] + invc * a2[t] + sc[(size_t)gn * 512 + col];
      }
    } else {
      const int i = (T - 8) >> 3, j = (T - 8) & 7, cb = j * 16 + n16;
      const float ry1 = 0.70710678118654752f * sY[n16 * 4 + 1 + i];
      const float ry0 = sC0[n16];
      const float* sVi = sV + (i * 16) * 128;
      // Three independent chains: vecs_lo@Wl1_top, vecs_hi@Wl1_bot, out_v@Wo1.
      v8f a1 = V8F_ZERO, a2 = V8F_ZERO, a3 = V8F_ZERO;
#pragma unroll 4
      for (int k0 = 0; k0 < 128; k0 += 4) {
        const int kk = k0 + kh;
        const float vi0 = sVi[n16 * 128 + kk];
        const float vi1 = sVi[n16 * 128 + kk + 1];
        v2f a, b;
        a[0] = ry1 * sW23[n16 * 256 + kk] * sS[n16 * 128 + kk];
        a[1] = ry1 * sW23[n16 * 256 + kk + 1] * sS[n16 * 128 + kk + 1];
        b[0] = Wl1[kk * 128 + cb];
        b[1] = Wl1[(kk + 1) * 128 + cb];
        a1 = wmma4(a1, a, b);
        a[0] = ry0 * sW23[n16 * 256 + 128 + kk] * vi0;
        a[1] = ry0 * sW23[n16 * 256 + 128 + kk + 1] * vi1;
        b[0] = Wl1[(128 + kk) * 128 + cb];
        b[1] = Wl1[(128 + kk + 1) * 128 + cb];
        a3 = wmma4(a3, a, b);
        a[0] = vi0;
        a[1] = vi1;
        b[0] = Wo1[kk * 128 + cb];
        b[1] = Wo1[(kk + 1) * 128 + cb];
        a2 = wmma4(a2, a, b);
      }
#pragma unroll
      for (int t = 0; t < 8; ++t) {
        const int m = t + 8 * rh, gn = base + m;
        const int o = j * 16 + n16, col = 128 + 3 * o + i;
        out[(size_t)gn * 512 + col] =
            inv2c * (a1[t] + a3[t]) + invc * a2[t] + sc[(size_t)gn * 512 + col];
      }
    }
  }
}

extern "C" void kernel_launch(void* const* d_in, const int* in_sizes, int n_in,
                              void* d_out, int out_size, void* d_ws, size_t ws_size,
                              hipStream_t stream) {
  const float* s_in  = (const float*)d_in[0];
  const float* v_in  = (const float*)d_in[1];
  const float* sc    = (const float*)d_in[2];
  const float* attrs = (const float*)d_in[3];
  const float* invf  = (const float*)d_in[4];
  const float* matt  = (const float*)d_in[5];
  const float* mlen  = (const float*)d_in[6];
  const float* Ws    = (const float*)d_in[7];
  const float* Wv    = (const float*)d_in[8];
  const float* Wc1   = (const float*)d_in[9];
  const float* Wc2   = (const float*)d_in[10];
  const float* Wc3   = (const float*)d_in[11];
  const float* Wc4   = (const float*)d_in[12];
  const float* Wb1   = (const float*)d_in[13];
  const float* Wb2   = (const float*)d_in[14];
  const float* Wb3   = (const float*)d_in[15];
  const float* Wb4   = (const float*)d_in[16];
  const float* esc   = (const float*)d_in[17];
  const float* Wl0   = (const float*)d_in[18];
  const float* Wl1   = (const float*)d_in[19];
  const float* Wo0   = (const float*)d_in[20];
  const float* Wo1   = (const float*)d_in[21];
  float* out = (float*)d_out;

  const int N = in_sizes[0] / 128;  // node count (C = 128)
  const int tiles = (N + 15) / 16;  // one 16-node tile per workgroup
  const size_t lds_bytes =
      (size_t)(2048 + 6144 + 4096 + 4096 + 1024 + 1024 + 256 + 64 + 16 + 16) *
      sizeof(float);  // ~73.4 KB dynamic LDS (WGP has 320 KB)

  magmom_fused<<<dim3(tiles), dim3(BLOCK), lds_bytes, stream>>>(
      s_in, v_in, sc, attrs, invf, matt, mlen, Ws, Wv, Wc1, Wc2, Wc3, Wc4, Wb1,
      Wb2, Wb3, Wb4, esc, Wl0, Wl1, Wo0, Wo1, out);
}